// Decoder_47811575939606
// MI455X (gfx1250) — compile-verified
//
#include <hip/hip_runtime.h>
#include <string.h>

// ---------------------------------------------------------------------------
// Pointer-network decoder for MI455X (gfx1250, wave32).
//  - e_g/e_p reference projections computed once with fp32 WMMA, stored bf16
//    (halves the ~63GB/step-loop HBM stream that dominates the roofline).
//  - All GEMMs use V_WMMA_F32_16X16X4_F32 (exact fp32 matrix path).
//  - Elementwise transcendentals use gfx1250 v_tanh/v_exp/v_log when available.
// ---------------------------------------------------------------------------

#define B_ 1024
#define S_ 200
#define E_ 256
#define H_ 256
#define NEG_INF (-__builtin_huge_valf())
#define L2E 1.4426950408889634f
#define LN2 0.6931471805599453f

typedef float v2f __attribute__((ext_vector_type(2)));
typedef float v8f __attribute__((ext_vector_type(8)));

// ---------------- fast math (gfx1250 TRANS ops when present) ----------------
__device__ __forceinline__ float fast_exp(float x) {
#if __has_builtin(__builtin_amdgcn_exp2f)
  return __builtin_amdgcn_exp2f(x * L2E);
#else
  return exp2f(x * L2E);
#endif
}
__device__ __forceinline__ float fast_log(float x) {
#if __has_builtin(__builtin_amdgcn_logf)
  return __builtin_amdgcn_logf(x) * LN2;
#else
  return log2f(x) * LN2;
#endif
}
__device__ __forceinline__ float fast_tanh(float x) {
#if __has_builtin(__builtin_amdgcn_tanhf)
  return __builtin_amdgcn_tanhf(x);
#elif __has_builtin(__builtin_amdgcn_tanh_f32)
  return __builtin_amdgcn_tanh_f32(x);
#else
  return 1.0f - 2.0f / (fast_exp(2.0f * x) + 1.0f);
#endif
}
__device__ __forceinline__ float sigmoidf_(float x) {
  return 1.0f / (1.0f + fast_exp(-x));
}
__device__ __forceinline__ unsigned short f2bf(float f) {  // RNE f32->bf16
  unsigned u = __float_as_uint(f);
  unsigned r = 0x7FFFu + ((u >> 16) & 1u);
  return (unsigned short)((u + r) >> 16);
}
__device__ __forceinline__ float bf2f(unsigned short b) {
  return __uint_as_float(((unsigned)b) << 16);
}

// ---------------------------------------------------------------------------
// WMMA fp32 GEMM:  C[m,n] = bias[n] + sum_k A0[m,k]*W0[n,k] + sum_k A1[m,k]*W1[n,k]
// A row-major [M,K], W row-major [N,K]. Wave computes a 16x64 tile.
// f32 16x16x4 fragment layout (ISA 7.12.2):
//   A: lane L holds A[tm + (L&15)][k0 + 2*(L>=16) + {0,1}]  -> contiguous v2f
//   B (=W^T): lane L holds W[tn + (L&15)][k0 + 2*(L>=16) + {0,1}] -> contiguous v2f
//   C: vgpr r, lane L -> row tm + r + 8*(L>=16), col tn + (L&15)
// ---------------------------------------------------------------------------
__global__ __launch_bounds__(128) void gemm_xwt_wmma(
    const float* __restrict__ A0, const float* __restrict__ W0, int K0,
    const float* __restrict__ A1, const float* __restrict__ W1, int K1,
    const float* __restrict__ bias, float* __restrict__ C, int M, int N) {
  const int lane = threadIdx.x & 31;
  const int wid  = blockIdx.x * (blockDim.x >> 5) + (threadIdx.x >> 5);
  const int tilesN = N >> 6;
  const int tm = (wid / tilesN) << 4;
  const int tn = (wid % tilesN) << 6;
  const int l15 = lane & 15;
  const int khalf = (lane >> 4) << 1;  // 0 or 2
  v8f acc0 = {}, acc1 = {}, acc2 = {}, acc3 = {};
  for (int seg = 0; seg < 2; ++seg) {
    const float* A = seg ? A1 : A0;
    const float* W = seg ? W1 : W0;
    const int K = seg ? K1 : K0;
    if (K == 0) continue;
    const float* ar = A + (size_t)(tm + l15) * K + khalf;
    const float* w0 = W + (size_t)(tn + l15) * K + khalf;
#pragma unroll 4
    for (int k = 0; k < K; k += 4) {
      v2f a  = *(const v2f*)(ar + k);
      v2f b0 = *(const v2f*)(w0 + k);
      v2f b1 = *(const v2f*)(w0 + (size_t)16 * K + k);
      v2f b2 = *(const v2f*)(w0 + (size_t)32 * K + k);
      v2f b3 = *(const v2f*)(w0 + (size_t)48 * K + k);
      acc0 = __builtin_amdgcn_wmma_f32_16x16x4_f32(false, a, false, b0, (short)0, acc0, false, false);
      acc1 = __builtin_amdgcn_wmma_f32_16x16x4_f32(false, a, false, b1, (short)0, acc1, false, false);
      acc2 = __builtin_amdgcn_wmma_f32_16x16x4_f32(false, a, false, b2, (short)0, acc2, false, false);
      acc3 = __builtin_amdgcn_wmma_f32_16x16x4_f32(false, a, false, b3, (short)0, acc3, false, false);
    }
  }
  const int rbase = tm + ((lane >> 4) << 3);
  const float bi0 = bias[tn + l15];
  const float bi1 = bias[tn + l15 + 16];
  const float bi2 = bias[tn + l15 + 32];
  const float bi3 = bias[tn + l15 + 48];
#pragma unroll
  for (int r = 0; r < 8; ++r) {
    float* crow = C + (size_t)(rbase + r) * N + tn + l15;
    crow[0]  = acc0[r] + bi0;
    crow[16] = acc1[r] + bi1;
    crow[32] = acc2[r] + bi2;
    crow[48] = acc3[r] + bi3;
  }
}

// ---------------------------------------------------------------------------
// Reference projection (once): E[b][s][o] = bf16( sum_h ctx[s][b][h]*Wref[o][h] + bref[o] )
// out rows r = b*S_+s  (M = B_*S_), cols o (N = H_). fp32 WMMA, bf16 store.
// ---------------------------------------------------------------------------
__global__ __launch_bounds__(128) void refproj_wmma(
    const float* __restrict__ ctx, const float* __restrict__ Wref,
    const float* __restrict__ bref, unsigned short* __restrict__ Ebf) {
  const int lane = threadIdx.x & 31;
  const int wid  = blockIdx.x * (blockDim.x >> 5) + (threadIdx.x >> 5);
  const int tilesN = H_ >> 6;  // 4
  const int tm = (wid / tilesN) << 4;
  const int tn = (wid % tilesN) << 6;
  const int l15 = lane & 15;
  const int khalf = (lane >> 4) << 1;
  const int r = tm + l15;
  const int b = r / S_, s = r % S_;
  const float* ar = ctx + ((size_t)s * B_ + b) * H_ + khalf;   // ctx is [S,B,H]
  const float* w0 = Wref + (size_t)(tn + l15) * H_ + khalf;
  v8f acc0 = {}, acc1 = {}, acc2 = {}, acc3 = {};
#pragma unroll 4
  for (int k = 0; k < H_; k += 4) {
    v2f a  = *(const v2f*)(ar + k);
    v2f b0 = *(const v2f*)(w0 + k);
    v2f b1 = *(const v2f*)(w0 + (size_t)16 * H_ + k);
    v2f b2 = *(const v2f*)(w0 + (size_t)32 * H_ + k);
    v2f b3 = *(const v2f*)(w0 + (size_t)48 * H_ + k);
    acc0 = __builtin_amdgcn_wmma_f32_16x16x4_f32(false, a, false, b0, (short)0, acc0, false, false);
    acc1 = __builtin_amdgcn_wmma_f32_16x16x4_f32(false, a, false, b1, (short)0, acc1, false, false);
    acc2 = __builtin_amdgcn_wmma_f32_16x16x4_f32(false, a, false, b2, (short)0, acc2, false, false);
    acc3 = __builtin_amdgcn_wmma_f32_16x16x4_f32(false, a, false, b3, (short)0, acc3, false, false);
  }
  const int rbase = tm + ((lane >> 4) << 3);
  const float bi0 = bref[tn + l15];
  const float bi1 = bref[tn + l15 + 16];
  const float bi2 = bref[tn + l15 + 32];
  const float bi3 = bref[tn + l15 + 48];
#pragma unroll
  for (int rr = 0; rr < 8; ++rr) {
    unsigned short* erow = Ebf + (size_t)(rbase + rr) * H_ + tn + l15;
    erow[0]  = f2bf(acc0[rr] + bi0);
    erow[16] = f2bf(acc1[rr] + bi1);
    erow[32] = f2bf(acc2[rr] + bi2);
    erow[48] = f2bf(acc3[rr] + bi3);
  }
}

// ------------------------- LSTM pointwise update ---------------------------
__global__ void lstm_elem(const float* __restrict__ gates, float* __restrict__ h,
                          float* __restrict__ c) {
  int i = blockIdx.x * blockDim.x + threadIdx.x;  // i < B_*H_
  int b = i >> 8, hh = i & 255;
  const float* g = gates + (size_t)b * (4 * H_);
  float ig = sigmoidf_(g[hh]);
  float fg = sigmoidf_(g[H_ + hh]);
  float gg = fast_tanh(g[2 * H_ + hh]);
  float og = sigmoidf_(g[3 * H_ + hh]);
  float cn = fg * c[i] + ig * gg;
  c[i] = cn;
  h[i] = og * fast_tanh(cn);
}

// --------- attention logits: u[b,s] = sum_h v[h]*tanh(E[b,s,h]+qp[b,h]) -----
// one wave per (b,s); lane covers 8 h values (16B coalesced bf16 load).
__global__ __launch_bounds__(256) void attn_logits_kernel(
    const unsigned short* __restrict__ E, const float* __restrict__ qp,
    const float* __restrict__ v, float* __restrict__ u) {
  const int lane = threadIdx.x & 31;
  const int gid = (blockIdx.x * blockDim.x + threadIdx.x) >> 5;  // = b*S_+s
  const int b = gid / S_;
  union { uint4 v4; unsigned short s[8]; } ld;
  ld.v4 = *(const uint4*)(E + (size_t)gid * H_ + lane * 8);
  const float* q  = qp + (size_t)b * H_ + lane * 8;
  const float* vp = v + lane * 8;
  float sum = 0.f;
#pragma unroll
  for (int t = 0; t < 8; ++t)
    sum += vp[t] * fast_tanh(bf2f(ld.s[t]) + q[t]);
#pragma unroll
  for (int off = 16; off; off >>= 1) sum += __shfl_down(sum, off, 32);
  if (lane == 0) u[gid] = sum;
}

// --------------- masked softmax over S per batch row (glimpse) -------------
__global__ __launch_bounds__(256) void masked_softmax(
    const float* __restrict__ u, const unsigned char* __restrict__ mask,
    float* __restrict__ w) {
  __shared__ float red[256];
  const int b = blockIdx.x, t = threadIdx.x;
  bool valid = (t < S_) && !mask[b * S_ + t];
  float x = valid ? u[b * S_ + t] : NEG_INF;
  red[t] = x; __syncthreads();
#pragma unroll
  for (int off = 128; off; off >>= 1) {
    if (t < off) red[t] = fmaxf(red[t], red[t + off]);
    __syncthreads();
  }
  float mx = red[0]; __syncthreads();
  float e = valid ? fast_exp(x - mx) : 0.f;
  red[t] = e; __syncthreads();
#pragma unroll
  for (int off = 128; off; off >>= 1) {
    if (t < off) red[t] += red[t + off];
    __syncthreads();
  }
  if (t < S_) w[b * S_ + t] = e / red[0];
}

// ------------- glimpse readout: gl[b,h] = sum_s E[b,s,h]*w[b,s] ------------
__global__ __launch_bounds__(256) void glimpse_readout(
    const unsigned short* __restrict__ Eg, const float* __restrict__ w,
    float* __restrict__ gl) {
  __shared__ float ws_[S_];
  const int b = blockIdx.x, t = threadIdx.x;  // t = h, 256 threads
  if (t < S_) ws_[t] = w[b * S_ + t];
  __syncthreads();
  const unsigned short* base = Eg + (size_t)b * S_ * H_ + t;
  float acc = 0.f;
#pragma unroll 4
  for (int s = 0; s < S_; ++s) acc += ws_[s] * bf2f(base[(size_t)s * H_]);
  gl[(size_t)b * H_ + t] = acc;
}

// ---- pointer head: 10*tanh(u), mask, log_softmax, argmax, gather next x ----
__global__ __launch_bounds__(256) void pointer_finish(
    const float* __restrict__ u, unsigned char* __restrict__ mask,
    const float* __restrict__ emb /*[S,B,E]*/, float* __restrict__ out_logp,
    float* __restrict__ out_sel, float* __restrict__ dec_in, int step) {
  __shared__ float red[256];
  __shared__ int redi[256];
  __shared__ int ssel;
  const int b = blockIdx.x, t = threadIdx.x;
  bool inS = (t < S_);
  bool valid = inS && !mask[b * S_ + t];
  float x = valid ? 10.0f * fast_tanh(u[b * S_ + t]) : NEG_INF;
  red[t] = x; redi[t] = t; __syncthreads();
#pragma unroll
  for (int off = 128; off; off >>= 1) {
    if (t < off) {
      float a = red[t], bb = red[t + off];
      int ia = redi[t], ib = redi[t + off];
      if (bb > a || (bb == a && ib < ia)) { red[t] = bb; redi[t] = ib; }
    }
    __syncthreads();
  }
  float mx = red[0];
  if (t == 0) ssel = redi[0];
  __syncthreads();
  float e = valid ? fast_exp(x - mx) : 0.f;
  red[t] = e; __syncthreads();
#pragma unroll
  for (int off = 128; off; off >>= 1) {
    if (t < off) red[t] += red[t + off];
    __syncthreads();
  }
  float logZ = fast_log(red[0]);
  if (inS)
    out_logp[((size_t)b * S_ + step) * S_ + t] = valid ? (x - mx - logZ) : NEG_INF;
  const int idx = ssel;
  if (t == 0) {
    out_sel[(size_t)b * S_ + step] = (float)idx;
    mask[b * S_ + idx] = 1;
  }
  // next decoder input: embedded_inputs[idx, b, :]  (t spans E_ == 256)
  dec_in[(size_t)b * E_ + t] = emb[((size_t)idx * B_ + b) * E_ + t];
}

// ------------------------------- small utils -------------------------------
__global__ void zero_bytes(unsigned char* p, int n) {
  int i = blockIdx.x * blockDim.x + threadIdx.x;
  if (i < n) p[i] = 0;
}
__global__ void bias_sum(const float* a, const float* b, float* o, int n) {
  int i = blockIdx.x * blockDim.x + threadIdx.x;
  if (i < n) o[i] = a[i] + b[i];
}

// ---------------------------------------------------------------------------
extern "C" void kernel_launch(void* const* d_in, const int* in_sizes, int n_in,
                              void* d_out, int out_size, void* d_ws, size_t ws_size,
                              hipStream_t stream) {
  const float* decoder_input = (const float*)d_in[0];
  const float* embedded      = (const float*)d_in[1];
  const float* h0    = (const float*)d_in[2];
  const float* c0    = (const float*)d_in[3];
  const float* ctx   = (const float*)d_in[4];
  const float* W_ih  = (const float*)d_in[5];
  const float* W_hh  = (const float*)d_in[6];
  const float* b_ih  = (const float*)d_in[7];
  const float* b_hh  = (const float*)d_in[8];
  const float* Wq_g  = (const float*)d_in[9];
  const float* bq_g  = (const float*)d_in[10];
  const float* Wref_g= (const float*)d_in[11];
  const float* bref_g= (const float*)d_in[12];
  const float* v_g   = (const float*)d_in[13];
  const float* Wq_p  = (const float*)d_in[14];
  const float* bq_p  = (const float*)d_in[15];
  const float* Wref_p= (const float*)d_in[16];
  const float* bref_p= (const float*)d_in[17];
  const float* v_p   = (const float*)d_in[18];

  // ---- workspace carve (aligned to 256B) ----
  char* ws = (char*)d_ws;
  size_t off = 0;
  auto carve = [&](size_t bytes) { void* p = ws + off; off += (bytes + 255) & ~(size_t)255; return p; };
  unsigned short* e_g = (unsigned short*)carve((size_t)B_ * S_ * H_ * 2);
  unsigned short* e_p = (unsigned short*)carve((size_t)B_ * S_ * H_ * 2);
  float* gates = (float*)carve((size_t)B_ * 4 * H_ * 4);
  float* h     = (float*)carve((size_t)B_ * H_ * 4);
  float* c     = (float*)carve((size_t)B_ * H_ * 4);
  float* dec   = (float*)carve((size_t)B_ * E_ * 4);
  float* qp    = (float*)carve((size_t)B_ * H_ * 4);
  float* gl    = (float*)carve((size_t)B_ * H_ * 4);
  float* u     = (float*)carve((size_t)B_ * S_ * 4);
  float* w     = (float*)carve((size_t)B_ * S_ * 4);
  float* bsum  = (float*)carve((size_t)4 * H_ * 4);
  unsigned char* mask = (unsigned char*)carve((size_t)B_ * S_);

  float* out      = (float*)d_out;
  float* out_logp = out;                               // [B, S, S]
  float* out_sel  = out + (size_t)B_ * S_ * S_;        // [B, S]
  float* out_h    = out_sel + (size_t)B_ * S_;         // [B, H]
  float* out_c    = out_h + (size_t)B_ * H_;           // [B, H]

  // ---- init state ----
  hipMemcpyAsync(h, h0, (size_t)B_ * H_ * 4, hipMemcpyDeviceToDevice, stream);
  hipMemcpyAsync(c, c0, (size_t)B_ * H_ * 4, hipMemcpyDeviceToDevice, stream);
  hipMemcpyAsync(dec, decoder_input, (size_t)B_ * E_ * 4, hipMemcpyDeviceToDevice, stream);
  zero_bytes<<<(B_ * S_ + 255) / 256, 256, 0, stream>>>(mask, B_ * S_);
  bias_sum<<<(4 * H_) / 256, 256, 0, stream>>>(b_ih, b_hh, bsum, 4 * H_);

  // ---- one-time reference projections (fp32 WMMA -> bf16) ----
  {
    int blocks = (B_ * S_ / 16) * (H_ / 64) / 4;  // 12800 blocks of 4 waves
    refproj_wmma<<<blocks, 128, 0, stream>>>(ctx, Wref_g, bref_g, e_g);
    refproj_wmma<<<blocks, 128, 0, stream>>>(ctx, Wref_p, bref_p, e_p);
  }

  const int gatesBlocks = (B_ / 16) * ((4 * H_) / 64) / 4;  // 256
  const int qpBlocks    = (B_ / 16) * (H_ / 64) / 4;        // 64
  const int attnBlocks  = (B_ * S_) / 8;                    // 25600 (8 waves/blk)

  for (int step = 0; step < S_; ++step) {
    // LSTM: gates = dec@W_ih^T + h@W_hh^T + (b_ih+b_hh)
    gemm_xwt_wmma<<<gatesBlocks, 128, 0, stream>>>(dec, W_ih, E_, h, W_hh, H_,
                                                   bsum, gates, B_, 4 * H_);
    lstm_elem<<<(B_ * H_) / 256, 256, 0, stream>>>(gates, h, c);
    // glimpse attention
    gemm_xwt_wmma<<<qpBlocks, 128, 0, stream>>>(h, Wq_g, H_, nullptr, nullptr, 0,
                                                bq_g, qp, B_, H_);
    attn_logits_kernel<<<attnBlocks, 256, 0, stream>>>(e_g, qp, v_g, u);
    masked_softmax<<<B_, 256, 0, stream>>>(u, mask, w);
    glimpse_readout<<<B_, 256, 0, stream>>>(e_g, w, gl);
    // pointer head
    gemm_xwt_wmma<<<qpBlocks, 128, 0, stream>>>(gl, Wq_p, H_, nullptr, nullptr, 0,
                                                bq_p, qp, B_, H_);
    attn_logits_kernel<<<attnBlocks, 256, 0, stream>>>(e_p, qp, v_p, u);
    pointer_finish<<<B_, 256, 0, stream>>>(u, mask, embedded, out_logp, out_sel,
                                           dec, step);
  }

  hipMemcpyAsync(out_h, h, (size_t)B_ * H_ * 4, hipMemcpyDeviceToDevice, stream);
  hipMemcpyAsync(out_c, c, (size_t)B_ * H_ * 4, hipMemcpyDeviceToDevice, stream);
}